// Routing_4252017623394
// MI455X (gfx1250) — compile-verified
//
#include <hip/hip_runtime.h>
#include <math.h>

typedef float v2f __attribute__((ext_vector_type(2)));
typedef float v4f __attribute__((ext_vector_type(4)));
typedef float v8f __attribute__((ext_vector_type(8)));

#define BATCH 8
#define NCAP 2048
#define DIN 256
#define RDIM 512
#define A_STRIDE 260   // pad: bank stride 4 -> 16 rows hit 16 distinct banks; 16B aligned rows

// ---------------------------------------------------------------------------
// Kernel 1: u_hat[b][n][r] = sum_d u[b][n][d] * W[n][d][r]
// One block (256 thr = 8 waves) per capsule n. M padded 8->16 for WMMA.
// fp32 WMMA (V_WMMA_F32_16X16X4_F32) keeps full precision; kernel is
// memory-bound on W (1.07 GB), so the 2x M-padding waste is free.
// ---------------------------------------------------------------------------
__global__ __launch_bounds__(256) void uhat_gemm(const float* __restrict__ u,
                                                 const float* __restrict__ W,
                                                 float* __restrict__ uhat) {
  __shared__ float Asm[16 * A_STRIDE];
  const int n = blockIdx.x;
  const int t = threadIdx.x;

  // Stage A tile (16 x 256): rows 0..7 = u[b][n][:], rows 8..15 = 0.
  {
    const int row = t >> 4;          // 0..15
    const int c0  = (t & 15) << 4;   // 0,16,...,240
    float* dst = &Asm[row * A_STRIDE + c0];
    if (row < BATCH) {
      const float* src = u + ((size_t)row * NCAP + n) * DIN + c0;
      #pragma unroll
      for (int q = 0; q < 4; ++q)
        *(v4f*)(dst + 4 * q) = *(const v4f*)(src + 4 * q);
    } else {
      v4f z = {0.f, 0.f, 0.f, 0.f};
      #pragma unroll
      for (int q = 0; q < 4; ++q)
        *(v4f*)(dst + 4 * q) = z;
    }
  }
  __syncthreads();

  const int wave = t >> 5;
  const int lane = t & 31;
  const int klo  = (lane < 16) ? 0 : 2;  // K-half held by this lane half

  // Per-lane pointers.
  // A operand (16x4 f32): lane holds M=lane&15, K = {k+klo, k+klo+1}.
  // B operand (4x16 f32): lane holds N=lane&15, K = {k+klo, k+klo+1}.
  const float* pA = &Asm[(lane & 15) * A_STRIDE + klo];
  const float* pB = W + ((size_t)n * DIN + klo) * RDIM + wave * 64 + (lane & 15);

  v8f acc0 = {0.f,0.f,0.f,0.f,0.f,0.f,0.f,0.f};
  v8f acc1 = acc0, acc2 = acc0, acc3 = acc0;

  #pragma unroll 4
  for (int k = 0; k < DIN; k += 4) {
    v2f a = *(const v2f*)(pA + k);
    v2f b0, b1, b2, b3;
    b0.x = pB[0];   b0.y = pB[RDIM];
    b1.x = pB[16];  b1.y = pB[RDIM + 16];
    b2.x = pB[32];  b2.y = pB[RDIM + 32];
    b3.x = pB[48];  b3.y = pB[RDIM + 48];
    acc0 = __builtin_amdgcn_wmma_f32_16x16x4_f32(false, a, false, b0, (short)0, acc0, false, false);
    acc1 = __builtin_amdgcn_wmma_f32_16x16x4_f32(false, a, false, b1, (short)0, acc1, false, false);
    acc2 = __builtin_amdgcn_wmma_f32_16x16x4_f32(false, a, false, b2, (short)0, acc2, false, false);
    acc3 = __builtin_amdgcn_wmma_f32_16x16x4_f32(false, a, false, b3, (short)0, acc3, false, false);
    pB += 4 * RDIM;
  }

  // C/D layout: VGPR j, lanes 0-15 -> M=j (valid b), N=lane. Lanes 16-31 hold
  // padded rows M=j+8 -> dropped.
  if (lane < 16) {
    const int cb = wave * 64 + lane;
    #pragma unroll
    for (int j = 0; j < BATCH; ++j) {
      float* o = uhat + ((size_t)j * NCAP + n) * RDIM + cb;
      o[0]  = acc0[j];
      o[16] = acc1[j];
      o[32] = acc2[j];
      o[48] = acc3[j];
    }
  }
}

// ---------------------------------------------------------------------------
// Kernel 2: one routing pass.
//   logit[b][n] = dot(Oprev[b], u_hat[b][n])   (or c=0.5 when use_prev==0)
//   c = sigmoid(logit)
//   P[b][chunk][r] = sum over this block's 64 n-rows of c * u_hat[b][n][r]
// Deterministic (no FP atomics): per-wave LDS partials + tree reduce.
// u_hat (33.5 MB) is L2-resident (192 MB L2).
// ---------------------------------------------------------------------------
__global__ __launch_bounds__(256) void routing_pass(const float* __restrict__ uhat,
                                                    const float* __restrict__ Oprev,
                                                    float* __restrict__ P,
                                                    int use_prev) {
  __shared__ float Sloc[8 * RDIM];
  const int b     = blockIdx.y;
  const int chunk = blockIdx.x;
  const int t     = threadIdx.x;
  const int wave  = t >> 5;
  const int lane  = t & 31;
  const int r0    = lane << 4;   // 16 r-columns per lane

  v4f o[4];
  #pragma unroll
  for (int q = 0; q < 4; ++q) { v4f z = {0.f,0.f,0.f,0.f}; o[q] = z; }
  if (use_prev) {
    const float* op = Oprev + b * RDIM + r0;
    #pragma unroll
    for (int q = 0; q < 4; ++q) o[q] = *(const v4f*)(op + 4 * q);
  }

  float acc[16];
  #pragma unroll
  for (int q = 0; q < 16; ++q) acc[q] = 0.f;

  const int nbase = chunk * 64 + wave * 8;
  for (int j = 0; j < 8; ++j) {
    const float* row = uhat + ((size_t)b * NCAP + (nbase + j)) * RDIM + r0;
    v4f x[4];
    #pragma unroll
    for (int q = 0; q < 4; ++q) x[q] = *(const v4f*)(row + 4 * q);

    float c;
    if (use_prev) {
      float p = 0.f;
      #pragma unroll
      for (int q = 0; q < 4; ++q)
        p += x[q].x * o[q].x + x[q].y * o[q].y + x[q].z * o[q].z + x[q].w * o[q].w;
      // wave32 cross-lane reduce
      #pragma unroll
      for (int off = 16; off > 0; off >>= 1)
        p += __shfl_xor(p, off, 32);
      c = 1.f / (1.f + expf(-p));
    } else {
      c = 0.5f;   // sigmoid(0)
    }

    #pragma unroll
    for (int q = 0; q < 4; ++q) {
      acc[4 * q + 0] += c * x[q].x;
      acc[4 * q + 1] += c * x[q].y;
      acc[4 * q + 2] += c * x[q].z;
      acc[4 * q + 3] += c * x[q].w;
    }
  }

  // Per-wave partials -> LDS (disjoint regions, deterministic)
  float* sl = &Sloc[wave * RDIM + r0];
  #pragma unroll
  for (int q = 0; q < 16; ++q) sl[q] = acc[q];
  __syncthreads();

  float s0 = 0.f, s1 = 0.f;
  #pragma unroll
  for (int w = 0; w < 8; ++w) {
    s0 += Sloc[w * RDIM + t];
    s1 += Sloc[w * RDIM + t + 256];
  }
  float* out = P + ((size_t)b * 32 + chunk) * RDIM;
  out[t]       = s0;
  out[t + 256] = s1;
}

// ---------------------------------------------------------------------------
// Kernel 3: fold the 32 per-chunk partials, then squash: x / sqrt(sum(x^2)+eps)
// ---------------------------------------------------------------------------
__global__ __launch_bounds__(256) void squash_reduce(const float* __restrict__ P,
                                                     float* __restrict__ O) {
  __shared__ float red[256];
  const int b = blockIdx.x;
  const int t = threadIdx.x;

  float s0 = 0.f, s1 = 0.f;
  for (int c = 0; c < 32; ++c) {
    const float* p = P + ((size_t)b * 32 + c) * RDIM;
    s0 += p[t];
    s1 += p[t + 256];
  }
  red[t] = s0 * s0 + s1 * s1;
  __syncthreads();
  for (int off = 128; off > 0; off >>= 1) {
    if (t < off) red[t] += red[t + off];
    __syncthreads();
  }
  const float inv = 1.f / sqrtf(red[0] + 1e-7f);
  O[b * RDIM + t]       = s0 * inv;
  O[b * RDIM + t + 256] = s1 * inv;
}

// ---------------------------------------------------------------------------
extern "C" void kernel_launch(void* const* d_in, const int* in_sizes, int n_in,
                              void* d_out, int out_size, void* d_ws, size_t ws_size,
                              hipStream_t stream) {
  const float* u = (const float*)d_in[0];   // (8, 2048, 256) f32
  const float* W = (const float*)d_in[1];   // (2048, 256, 512) f32
  float* out = (float*)d_out;               // (8, 1, 512) f32

  char* ws = (char*)d_ws;
  float* uhat = (float*)ws;                                         // 33,554,432 B
  float* P    = (float*)(ws + (size_t)BATCH * NCAP * RDIM * 4);     //    524,288 B
  float* O0   = P  + (size_t)BATCH * 32 * RDIM;                     //     16,384 B
  float* O1   = O0 + (size_t)BATCH * RDIM;                          //     16,384 B

  uhat_gemm<<<NCAP, 256, 0, stream>>>(u, W, uhat);

  // iter 0: c = sigmoid(0) = 0.5
  routing_pass<<<dim3(32, BATCH), 256, 0, stream>>>(uhat, O0, P, 0);
  squash_reduce<<<BATCH, 256, 0, stream>>>(P, O0);
  // iter 1
  routing_pass<<<dim3(32, BATCH), 256, 0, stream>>>(uhat, O0, P, 1);
  squash_reduce<<<BATCH, 256, 0, stream>>>(P, O1);
  // iter 2 (final -> d_out)
  routing_pass<<<dim3(32, BATCH), 256, 0, stream>>>(uhat, O1, P, 1);
  squash_reduce<<<BATCH, 256, 0, stream>>>(P, out);
}